// GLiftingKernel_8753143349717
// MI455X (gfx1250) — compile-verified
//
#include <hip/hip_runtime.h>
#include <hip/hip_bf16.h>

typedef __attribute__((ext_vector_type(2))) float v2f;
typedef __attribute__((ext_vector_type(8))) float v8f;
typedef __attribute__((ext_vector_type(4))) unsigned int v4u;
typedef __attribute__((ext_vector_type(8))) int v8i_;
typedef __attribute__((ext_vector_type(4))) int v4i_;

#define NUM_H 16
#define KS    7
#define P49   49          // ks*ks
#define M_ROWS 784        // NUM_H * P49
#define K_PAD  52         // 49 padded to multiple of 4
#define K_TILES 13
#define M_TILES 49        // 784 / 16
#define CIN   256
#define COUT  256
#define A_ELEMS (M_ROWS * K_PAD)   // 40768 floats = 163072 B

// ---------------------------------------------------------------------------
// Kernel 1: build the 784x52 interpolation matrix A into d_ws.
// A[m][q] = mask[p]/|det(H_h)| * bilinear coefficient of weight tap q for
// output row m = h*49 + p.  4 nonzeros per row (accumulated, so border-clip
// coincidences are handled like the reference's sum of corner terms).
// ---------------------------------------------------------------------------
__global__ void glift_build_A(const float* __restrict__ Hm,
                              const float* __restrict__ grid,
                              const float* __restrict__ mask,
                              float* __restrict__ A) {
  int m = blockIdx.x * blockDim.x + threadIdx.x;
  if (m >= M_ROWS) return;
  int h = m / P49;
  int p = m - h * P49;

  float H00 = Hm[h * 4 + 0], H01 = Hm[h * 4 + 1];
  float H10 = Hm[h * 4 + 2], H11 = Hm[h * 4 + 3];
  float det = fabsf(H00 * H11 - H01 * H10);

  float gx = grid[p * 2 + 0];
  float gy = grid[p * 2 + 1];
  // H_inv = H^T ; coord_i = sum_j Hinv[i][j] * g[j]
  float xr = H00 * gx + H10 * gy;
  float yr = H01 * gx + H11 * gy;

  float x = (xr + 1.0f) * 0.5f * (float)(KS - 1);
  float y = (yr + 1.0f) * 0.5f * (float)(KS - 1);
  float x0f = floorf(x), y0f = floorf(y);
  float wx = x - x0f, wy = y - y0f;
  int x0 = min(max((int)x0f, 0), KS - 1);
  int x1 = min(x0 + 1, KS - 1);
  int y0 = min(max((int)y0f, 0), KS - 1);
  int y1 = min(y0 + 1, KS - 1);

  float s = mask[p] / det;

  float row[K_PAD];
#pragma unroll
  for (int q = 0; q < K_PAD; ++q) row[q] = 0.0f;
  row[y0 * KS + x0] += s * (1.0f - wx) * (1.0f - wy);
  row[y0 * KS + x1] += s * wx * (1.0f - wy);
  row[y1 * KS + x0] += s * (1.0f - wx) * wy;
  row[y1 * KS + x1] += s * wx * wy;

  for (int q = 0; q < K_PAD; ++q) A[m * K_PAD + q] = row[q];
}

// ---------------------------------------------------------------------------
// Kernel 2: OUT = A (784x52, LDS via TDM) x W (52x65536, registers per
// wave-N-tile) via V_WMMA_F32_16X16X4_F32.  The 205 MB output stream is the
// bottleneck (~8.8us at 23.3 TB/s); stores are branchless b32 with NT hint.
// ---------------------------------------------------------------------------
__global__ __launch_bounds__(256, 1)
void glift_wmma(const float* __restrict__ W,   // weight [n][49], n = co*256+ci
                const float* __restrict__ A,   // [784][52] from kernel 1
                float* __restrict__ out) {
  extern __shared__ float ldsA[];              // 784*52 floats = 163072 B

  int tid  = threadIdx.x;
  int wave = tid >> 5;
  int lane = tid & 31;
  int nl   = lane & 15;     // column inside N-tile
  int sel  = lane >> 4;     // hi/lo half selector (ISA A/B fragment layout)

#if defined(__gfx1250__) && __has_builtin(__builtin_amdgcn_tensor_load_to_lds)
  // --- Tensor Data Mover: async-stage A (163 KB) into LDS with one D#. ---
  if (wave == 0) {
    unsigned long long ga = (unsigned long long)A;
    unsigned int lds_off = (unsigned int)(unsigned long long)(void*)ldsA;
    // D# group 0: count=1 | lds_addr | global_addr[56:0] | type=2
    v4u g0 = { 1u,                                  // count=1, rest 0
               lds_off,                             // bits 63:32 lds_addr
               (unsigned int)(ga & 0xffffffffull),  // bits 95:64
               (unsigned int)((ga >> 32) & 0x1ffffffull) | 0x80000000u };
    // D# group 1: data_size=4B; 1 row of 40768 elements (tile == tensor).
    v8i_ g1 = { (int)0x00020000u,                   // data_size=2 (4 bytes)
                (int)((A_ELEMS & 0xffff) << 16),    // tensor_dim0[15:0]
                (int)(1u << 16),                    // tensor_dim1 = 1
                (int)((unsigned)A_ELEMS << 16),     // tile_dim0 = 40768
                1,                                  // tile_dim1 = 1
                A_ELEMS,                            // tensor_dim0_stride
                (int)((unsigned)A_ELEMS << 16),     // tensor_dim1_stride[15:0]
                0 };
    v4i_ gz = { 0, 0, 0, 0 };
#if __clang_major__ >= 23
    v8i_ g4 = { 0, 0, 0, 0, 0, 0, 0, 0 };
    __builtin_amdgcn_tensor_load_to_lds(g0, g1, gz, gz, g4, 0);
#else
    __builtin_amdgcn_tensor_load_to_lds(g0, g1, gz, gz, 0);
#endif
    __builtin_amdgcn_s_wait_tensorcnt(0);
  }
#else
  // Fallback: cooperative float4 stage of A into LDS.
  {
    const float4* src = (const float4*)A;
    float4* dst = (float4*)ldsA;
    for (int i = tid; i < A_ELEMS / 4; i += 256) dst[i] = src[i];
  }
#endif

  int ntile = blockIdx.x * 8 + wave;
  int n     = ntile * 16 + nl;               // channel pair co*256+ci
  const float* wrow = W + n * P49;

  // B fragments: B[q][n], q = kt*4 + sel*2 + {0,1}; rows >= 49 are zero pad.
  // Loaded while the TDM streams A into LDS.
  v2f bfrag[K_TILES];
#pragma unroll
  for (int kt = 0; kt < K_TILES; ++kt) {
    int q0 = kt * 4 + sel * 2;
    bfrag[kt].x = (q0 < P49) ? wrow[q0] : 0.0f;
    bfrag[kt].y = (q0 + 1 < P49) ? wrow[q0 + 1] : 0.0f;
  }

  __syncthreads();   // A visible in LDS to all 8 waves

  int co = n >> 8;
  int ci = n & 255;
  // out[((co*16 + h)*256 + ci)*49 + p] = co*200704 + ci*49 + h*12544 + p
  float* outn = out + co * (NUM_H * CIN * P49) + ci * P49;

  for (int mt = 0; mt < M_TILES; ++mt) {
    v8f acc = {0.f, 0.f, 0.f, 0.f, 0.f, 0.f, 0.f, 0.f};
    const float* arow = ldsA + (mt * 16 + nl) * K_PAD + sel * 2;
#pragma unroll
    for (int kt = 0; kt < K_TILES; ++kt) {
      v2f a;
      a.x = arow[kt * 4];
      a.y = arow[kt * 4 + 1];
      acc = __builtin_amdgcn_wmma_f32_16x16x4_f32(
          /*neg_a=*/false, a, /*neg_b=*/false, bfrag[kt],
          /*c_mod=*/(short)0, acc, /*reuse_a=*/false, /*reuse_b=*/false);
    }
    // Lane's 8 accumulator values are rows m0..m0+7 (consecutive p) at fixed
    // n.  At most one h-row boundary inside the group -> branchless select
    // instead of exec-mask divergence; NT hint keeps the 205 MB stream from
    // thrashing L2 (which holds W + A).
    int m0 = mt * 16 + sel * 8;
    int h  = m0 / P49;
    int p0 = m0 - h * P49;
    float* dst = outn + h * (CIN * P49) + p0;
#pragma unroll
    for (int v = 0; v < 8; ++v) {
      int wrap = (p0 + v >= P49) ? (CIN * P49 - P49) : 0;
      __builtin_nontemporal_store(acc[v], dst + v + wrap);
    }
  }
}

// ---------------------------------------------------------------------------
extern "C" void kernel_launch(void* const* d_in, const int* in_sizes, int n_in,
                              void* d_out, int out_size, void* d_ws, size_t ws_size,
                              hipStream_t stream) {
  const float* Hm     = (const float*)d_in[0];  // [16,2,2]
  const float* weight = (const float*)d_in[1];  // [256,256,7,7]
  const float* grid   = (const float*)d_in[2];  // [7,7,2]
  const float* mask   = (const float*)d_in[3];  // [7,7]
  float* A   = (float*)d_ws;                    // 784*52 floats = 163072 B
  float* out = (float*)d_out;                   // [256,16,256,7,7]

  glift_build_A<<<4, 256, 0, stream>>>(Hm, grid, mask, A);

  const size_t ldsBytes = (size_t)A_ELEMS * sizeof(float); // 163072
  glift_wmma<<<(CIN * COUT) / (16 * 8), 256, ldsBytes, stream>>>(weight, A, out);
}